// WeightedRigidAlign_81372450390572
// MI455X (gfx1250) — compile-verified
//
#include <hip/hip_runtime.h>
#include <stdint.h>

// Weighted rigid (Kabsch) alignment for B=256 batches of N=16384 atoms.
// Pass 1: bandwidth-bound weighted moment reduction (~122 MB reads).
// Pass 2: per-batch 3x3 solve (f64 Jacobi on cov^T cov) + WMMA f32 16x16x4
//         computing rot = U * diag(1,1,flip) * V^T  (one wave per batch).
// Pass 3: bandwidth-bound affine apply (~52 MB read from L2 + 48 MB write).

#define BATCH   256
#define NATOMS  16384
#define SLICES  8
#define K1_THR  256
#define ATOMS_PER_THREAD ((NATOMS / SLICES) / K1_THR)   // 8

typedef float v2f __attribute__((ext_vector_type(2)));
typedef float v8f __attribute__((ext_vector_type(8)));

// ---------------------------------------------------------------- pass 1 ----
__global__ __launch_bounds__(K1_THR)
void wra_reduce(const float* __restrict__ pred, const float* __restrict__ tru,
                const float* __restrict__ wgt, const unsigned char* __restrict__ msk,
                float* __restrict__ part)
{
  const int s    = blockIdx.x;          // slice
  const int b    = blockIdx.y;          // batch
  const int tid  = threadIdx.x;
  const int lane = tid & 31;
  const int wave = tid >> 5;

  // acc: [0]=wsum [1..3]=sum w*t [4..6]=sum w*p [7..15]=sum w*t_i*p_j (row major)
  float acc[16];
#pragma unroll
  for (int v = 0; v < 16; ++v) acc[v] = 0.0f;

  const size_t baseAtom = (size_t)b * NATOMS + (size_t)s * (NATOMS / SLICES);
#pragma unroll
  for (int i = 0; i < ATOMS_PER_THREAD; ++i) {
    size_t off = baseAtom + (size_t)i * K1_THR + tid;
    float w = wgt[off];
    unsigned char mk = msk[off];
    w = mk ? w : 0.0f;                      // zeroing w zeroes every pass-1 sum
    const float* pp = pred + off * 3;
    const float* tp = tru  + off * 3;
    float px = pp[0], py = pp[1], pz = pp[2];
    float tx = tp[0], ty = tp[1], tz = tp[2];
    float wtx = w * tx, wty = w * ty, wtz = w * tz;
    acc[0] += w;
    acc[1] += wtx;      acc[2] += wty;      acc[3] += wtz;
    acc[4] += w * px;   acc[5] += w * py;   acc[6] += w * pz;
    acc[7]  = fmaf(wtx, px, acc[7]);  acc[8]  = fmaf(wtx, py, acc[8]);  acc[9]  = fmaf(wtx, pz, acc[9]);
    acc[10] = fmaf(wty, px, acc[10]); acc[11] = fmaf(wty, py, acc[11]); acc[12] = fmaf(wty, pz, acc[12]);
    acc[13] = fmaf(wtz, px, acc[13]); acc[14] = fmaf(wtz, py, acc[14]); acc[15] = fmaf(wtz, pz, acc[15]);
  }

  // wave32 butterfly reduction
#pragma unroll
  for (int v = 0; v < 16; ++v) {
    float x = acc[v];
    x += __shfl_xor(x, 16, 32);
    x += __shfl_xor(x,  8, 32);
    x += __shfl_xor(x,  4, 32);
    x += __shfl_xor(x,  2, 32);
    x += __shfl_xor(x,  1, 32);
    acc[v] = x;
  }

  __shared__ float sred[K1_THR / 32][16];
  if (lane == 0) {
#pragma unroll
    for (int v = 0; v < 16; ++v) sred[wave][v] = acc[v];
  }
  __syncthreads();
  if (tid < 16) {
    float tot = 0.0f;
#pragma unroll
    for (int w = 0; w < K1_THR / 32; ++w) tot += sred[w][tid];
    part[((size_t)b * SLICES + s) * 16 + tid] = tot;
  }
}

// ---------------------------------------------------------------- pass 2 ----
__device__ __forceinline__ float sel33f(int i, int j,
    float m00, float m01, float m02,
    float m10, float m11, float m12,
    float m20, float m21, float m22)
{
  float r0 = (j == 0) ? m00 : ((j == 1) ? m01 : m02);
  float r1 = (j == 0) ? m10 : ((j == 1) ? m11 : m12);
  float r2 = (j == 0) ? m20 : ((j == 1) ? m21 : m22);
  float r  = (i == 0) ? r0  : ((i == 1) ? r1  : r2);
  return (i < 3 && j < 3) ? r : 0.0f;
}

__global__ __launch_bounds__(32)
void wra_solve(const float* __restrict__ part, float* __restrict__ mrot)
{
  const int b    = blockIdx.x;
  const int lane = threadIdx.x;

  float sums[16];
#pragma unroll
  for (int v = 0; v < 16; ++v) sums[v] = 0.0f;
  for (int sI = 0; sI < SLICES; ++sI) {
#pragma unroll
    for (int v = 0; v < 16; ++v)
      sums[v] += part[((size_t)b * SLICES + sI) * 16 + v];
  }

  // cov[i][j] = S_ij - St_i * Sp_j / wsum  (replicated in every lane, f64)
  double wsum = (double)sums[0];
  double st0 = sums[1], st1 = sums[2], st2 = sums[3];
  double sp0 = sums[4], sp1 = sums[5], sp2 = sums[6];
  double inw = 1.0 / wsum;
  double c00 = sums[7]  - st0 * sp0 * inw;
  double c01 = sums[8]  - st0 * sp1 * inw;
  double c02 = sums[9]  - st0 * sp2 * inw;
  double c10 = sums[10] - st1 * sp0 * inw;
  double c11 = sums[11] - st1 * sp1 * inw;
  double c12 = sums[12] - st1 * sp2 * inw;
  double c20 = sums[13] - st2 * sp0 * inw;
  double c21 = sums[14] - st2 * sp1 * inw;
  double c22 = sums[15] - st2 * sp2 * inw;

  // A = cov^T cov (symmetric)
  double a00 = c00*c00 + c10*c10 + c20*c20;
  double a01 = c00*c01 + c10*c11 + c20*c21;
  double a02 = c00*c02 + c10*c12 + c20*c22;
  double a11 = c01*c01 + c11*c11 + c21*c21;
  double a12 = c01*c02 + c11*c12 + c21*c22;
  double a22 = c02*c02 + c12*c12 + c22*c22;

  double v[3][3] = {{1,0,0},{0,1,0},{0,0,1}};

  // Branchless cyclic Jacobi rotation; all lanes hold identical data so
  // EXEC stays fully converged at the WMMA below.
#define JROT(APP,AQQ,APQ,ARP,ARQ,P,Q)  do {                                   \
    double apq_ = (APQ);                                                      \
    double th_  = ((AQQ) - (APP)) * 0.5 / apq_;                               \
    double t_   = copysign(1.0, th_) / (fabs(th_) + sqrt(th_*th_ + 1.0));     \
    t_ = (fabs(apq_) > 1e-300) ? t_ : 0.0;                                    \
    double cc_ = 1.0 / sqrt(t_*t_ + 1.0);                                     \
    double ss_ = t_ * cc_;                                                    \
    (APP) -= t_ * apq_; (AQQ) += t_ * apq_; (APQ) = 0.0;                      \
    double rp_ = (ARP), rq_ = (ARQ);                                          \
    (ARP) = cc_*rp_ - ss_*rq_; (ARQ) = ss_*rp_ + cc_*rq_;                     \
    _Pragma("unroll")                                                         \
    for (int i_ = 0; i_ < 3; ++i_) {                                          \
      double vp_ = v[i_][P], vq_ = v[i_][Q];                                  \
      v[i_][P] = cc_*vp_ - ss_*vq_; v[i_][Q] = ss_*vp_ + cc_*vq_;             \
    }                                                                         \
  } while (0)

  for (int sweep = 0; sweep < 8; ++sweep) {
    JROT(a00, a11, a01, a02, a12, 0, 1);
    JROT(a00, a22, a02, a01, a12, 0, 2);
    JROT(a11, a22, a12, a01, a02, 1, 2);
  }
#undef JROT

  // sort eigenvalues descending, carry V columns (branchless)
  double l0 = a00, l1 = a11, l2 = a22;
#define CSW(LI,LJ,I,J) do {                                                   \
    bool sw_ = (LI) < (LJ);                                                   \
    double tl_ = sw_ ? (LJ) : (LI); (LJ) = sw_ ? (LI) : (LJ); (LI) = tl_;     \
    _Pragma("unroll")                                                         \
    for (int i_ = 0; i_ < 3; ++i_) {                                          \
      double ti_ = sw_ ? v[i_][J] : v[i_][I];                                 \
      double tj_ = sw_ ? v[i_][I] : v[i_][J];                                 \
      v[i_][I] = ti_; v[i_][J] = tj_;                                         \
    }                                                                         \
  } while (0)
  CSW(l0, l1, 0, 1); CSW(l0, l2, 0, 2); CSW(l1, l2, 1, 2);
#undef CSW

  double i0 = 1.0 / fmax(sqrt(fmax(l0, 0.0)), 1e-30);
  double i1 = 1.0 / fmax(sqrt(fmax(l1, 0.0)), 1e-30);
  double i2 = 1.0 / fmax(sqrt(fmax(l2, 0.0)), 1e-30);

  // U columns: u_k = cov * v_k / s_k
  float U00 = (float)((c00*v[0][0] + c01*v[1][0] + c02*v[2][0]) * i0);
  float U10 = (float)((c10*v[0][0] + c11*v[1][0] + c12*v[2][0]) * i0);
  float U20 = (float)((c20*v[0][0] + c21*v[1][0] + c22*v[2][0]) * i0);
  float U01 = (float)((c00*v[0][1] + c01*v[1][1] + c02*v[2][1]) * i1);
  float U11 = (float)((c10*v[0][1] + c11*v[1][1] + c12*v[2][1]) * i1);
  float U21 = (float)((c20*v[0][1] + c21*v[1][1] + c22*v[2][1]) * i1);
  float U02 = (float)((c00*v[0][2] + c01*v[1][2] + c02*v[2][2]) * i2);
  float U12 = (float)((c10*v[0][2] + c11*v[1][2] + c12*v[2][2]) * i2);
  float U22 = (float)((c20*v[0][2] + c21*v[1][2] + c22*v[2][2]) * i2);

  double det = c00*(c11*c22 - c12*c21) - c01*(c10*c22 - c12*c20)
             + c02*(c10*c21 - c11*c20);
  float flip = (det < 0.0) ? -1.0f : 1.0f;   // sign(det U V^T) since sigma>=0

  float V00=(float)v[0][0], V01=(float)v[0][1], V02=(float)v[0][2];
  float V10=(float)v[1][0], V11=(float)v[1][1], V12=(float)v[1][2];
  float V20=(float)v[2][0], V21=(float)v[2][1], V22=(float)v[2][2];

  // --- WMMA f32 16x16x4: rot = U * diag(1,1,flip) * V^T -------------------
  // 32-bit A 16x4 layout: lane m (0-15) K=0 in v0 / K=1 in v1; lane m+16 K=2/K=3.
  // B 4x16 mirrors it: lane n (0-15) holds row K=0 (v0) / K=1 (v1); +16 -> K=2/3.
  int m  = lane & 15;
  int pr = lane >> 4;
  int k0 = 2 * pr, k1 = k0 + 1;

  v2f Aop, Bop;
  Aop.x = sel33f(m, k0, U00,U01,U02, U10,U11,U12, U20,U21,U22);
  Aop.y = sel33f(m, k1, U00,U01,U02, U10,U11,U12, U20,U21,U22);
  // B[k][n] = flip_k * V[n][k]
  float f0 = (k0 == 2) ? flip : 1.0f;
  float f1 = (k1 == 2) ? flip : 1.0f;           // k1==3 zeroed by sel33f
  Bop.x = sel33f(m, k0, V00,V01,V02, V10,V11,V12, V20,V21,V22) * f0;
  Bop.y = sel33f(m, k1, V00,V01,V02, V10,V11,V12, V20,V21,V22) * f1;

  v8f Cop = {};
  Cop = __builtin_amdgcn_wmma_f32_16x16x4_f32(
      /*neg_a=*/false, Aop, /*neg_b=*/false, Bop,
      /*c_mod=*/(short)0, Cop, /*reuse_a=*/false, /*reuse_b=*/false);
  // D layout: lane j (0-15) holds rot[i][j] in Cop[i] for i=0..7.

  float pc0 = (float)(sp0 * inw), pc1 = (float)(sp1 * inw), pc2 = (float)(sp2 * inw);
  float tc0 = (float)(st0 * inw), tc1 = (float)(st1 * inw), tc2 = (float)(st2 * inw);

  if (lane < 3) {            // lane j emits affine column j
    float r0 = Cop[0], r1 = Cop[1], r2 = Cop[2];
    float tcj  = (lane == 0) ? tc0 : ((lane == 1) ? tc1 : tc2);
    float row3 = tcj - (pc0 * r0 + pc1 * r1 + pc2 * r2);
    *(float4*)(mrot + (size_t)b * 16 + lane * 4) = make_float4(r0, r1, r2, row3);
  }
}

// ---------------------------------------------------------------- pass 3 ----
__global__ __launch_bounds__(256)
void wra_apply(const float* __restrict__ pred, const unsigned char* __restrict__ msk,
               const float* __restrict__ mrot, float* __restrict__ out)
{
  size_t idx = (size_t)blockIdx.x * 256 + threadIdx.x;   // atom index over B*N
  int b = (int)(idx >> 14);                              // N = 16384
  const float4* M = (const float4*)(mrot + (size_t)b * 16);
  float4 c0 = M[0], c1 = M[1], c2 = M[2];

  unsigned char mk = msk[idx];
  const float* pp = pred + idx * 3;
  float px = pp[0], py = pp[1], pz = pp[2];
  px = mk ? px : 0.0f; py = mk ? py : 0.0f; pz = mk ? pz : 0.0f;

  float o0 = fmaf(px, c0.x, fmaf(py, c0.y, fmaf(pz, c0.z, c0.w)));
  float o1 = fmaf(px, c1.x, fmaf(py, c1.y, fmaf(pz, c1.z, c1.w)));
  float o2 = fmaf(px, c2.x, fmaf(py, c2.y, fmaf(pz, c2.z, c2.w)));

  float* op = out + idx * 3;
  op[0] = o0; op[1] = o1; op[2] = o2;
}

// ---------------------------------------------------------------- launch ----
extern "C" void kernel_launch(void* const* d_in, const int* in_sizes, int n_in,
                              void* d_out, int out_size, void* d_ws, size_t ws_size,
                              hipStream_t stream)
{
  (void)in_sizes; (void)n_in; (void)out_size; (void)ws_size;
  const float*         pred = (const float*)d_in[0];
  const float*         tru  = (const float*)d_in[1];
  const float*         wgt  = (const float*)d_in[2];
  const unsigned char* msk  = (const unsigned char*)d_in[3];  // jax bool = 1 byte
  float* out = (float*)d_out;

  float* part = (float*)d_ws;                          // [B][SLICES][16]  (128 KB)
  float* mrot = part + (size_t)BATCH * SLICES * 16;    // [B][16]          (16 KB)

  dim3 g1(SLICES, BATCH);
  wra_reduce<<<g1, K1_THR, 0, stream>>>(pred, tru, wgt, msk, part);
  wra_solve<<<BATCH, 32, 0, stream>>>(part, mrot);
  wra_apply<<<(BATCH * NATOMS) / 256, 256, 0, stream>>>(pred, msk, mrot, out);
}